// PointEdgeLengthLoss_1382979470104
// MI455X (gfx1250) — compile-verified
//
#include <hip/hip_runtime.h>

typedef __attribute__((ext_vector_type(2))) float v2f;
typedef __attribute__((ext_vector_type(8))) float v8f;

constexpr int Bb = 2;
constexpr int Nn = 8192;
constexpr int Kk = 16;            // neighbors kept (self excluded via diagonal mask)
constexpr float BIG = 3.0e38f;

// ---------------------------------------------------------------------------
// Kernel 1: squared norms of the reference cloud
// ---------------------------------------------------------------------------
__global__ void sq_kernel(const float* __restrict__ pref, float* __restrict__ sq) {
    int i = blockIdx.x * blockDim.x + threadIdx.x;
    if (i < Bb * Nn) {
        const float* p = pref + (size_t)i * 3;
        sq[i] = p[0] * p[0] + p[1] * p[1] + p[2] * p[2];
    }
}

// ---------------------------------------------------------------------------
// Kernel 2: fused WMMA gram-tile + top-16 selection.
// One wave (32 lanes) handles a block of 16 query rows; sweeps 512 column
// tiles using v_wmma_f32_16x16x4_f32 to produce 16x16 dot products.
//
// A (16x4 f32, MxK) per-lane layout (ISA 7.12.2):
//   lanes 0-15 : VGPR0 = K0(x), VGPR1 = K1(y), M = lane
//   lanes 16-31: VGPR0 = K2(z), VGPR1 = K3(0), M = lane-16
// B (4x16 f32, KxN) is the mirrored layout over columns.
// D (16x16 f32): VGPR r holds row r (lanes 0-15) / row r+8 (lanes 16-31),
// column = lane & 15.
//
// B-tile loads are software-pipelined one iteration ahead so they overlap
// the WMMA + LDS-staged selection of the current tile. Diagonal (self)
// masking is hoisted to the single uniform m0 == n0 tile.
// ---------------------------------------------------------------------------
__global__ void __launch_bounds__(32)
knn_kernel(const float* __restrict__ pref, const float* __restrict__ sq,
           int* __restrict__ knn) {
    __shared__ float tile[16][17];   // d2 tile, padded against bank conflicts
    __shared__ float md[16][32];     // merge buffers (dist)
    __shared__ int   mi[16][32];     // merge buffers (index)

    const int  lane  = threadIdx.x;
    const int  col   = lane & 15;
    const bool hi    = lane >= 16;
    const int  rbase = hi ? 8 : 0;   // D-matrix row base for this half-wave
    const int  b     = blockIdx.y;
    const int  n0    = blockIdx.x * 16;

    const float* pb  = pref + (size_t)b * Nn * 3;
    const float* sqb = sq   + (size_t)b * Nn;

    // A operand: query rows n0..n0+15 (loop invariant). Load all three
    // coordinates unconditionally; select with cndmask (no exec branching).
    {
        // nothing
    }
    const float* pa = pb + (size_t)(n0 + col) * 3;
    float ax = pa[0], ay = pa[1], az = pa[2];
    v2f a;
    a.x = hi ? az   : ax;
    a.y = hi ? 0.0f : ay;

    // ||p||^2 for the 8 rows this lane's D-registers cover
    float srow[8];
#pragma unroll
    for (int r = 0; r < 8; ++r) srow[r] = sqb[n0 + rbase + r];

    // Register-resident sorted top-16 (ascending) for row (lane&15);
    // half-wave L scans cols 0..7 of each tile, L+16 scans cols 8..15.
    float bd[Kk];
    int   bi[Kk];
#pragma unroll
    for (int k = 0; k < Kk; ++k) { bd[k] = BIG; bi[k] = 0; }

    const int myrow = col;
    const int cbase = rbase;

    // ---- software pipeline: preload B tile 0 ----
    const float* pc0 = pb + (size_t)col * 3;
    float bx = pc0[0], by = pc0[1], bz = pc0[2];
    float sc = sqb[col];

    for (int m0 = 0; m0 < Nn; m0 += 16) {
        // current B operand from pipelined registers
        v2f bm;
        bm.x = hi ? bz   : bx;
        bm.y = hi ? 0.0f : by;
        float scol = sc;

        // issue next tile's loads now (wrap to 0 on last iter; harmless)
        int mn = (m0 + 16 < Nn) ? (m0 + 16) : 0;
        const float* pc = pb + (size_t)(mn + col) * 3;
        float nbx = pc[0], nby = pc[1], nbz = pc[2];
        float nsc = sqb[mn + col];

        v8f c = {};
        c = __builtin_amdgcn_wmma_f32_16x16x4_f32(
                false, a, false, bm, (short)0, c, false, false);

        // d2 = ||pn||^2 + ||pm||^2 - 2 pn.pm
#pragma unroll
        for (int r = 0; r < 8; ++r) {
            float d2 = srow[r] + scol - 2.0f * c[r];
            tile[rbase + r][col] = d2;
        }
        // self-distance can only appear in the one aligned diagonal tile
        if (m0 == n0) {
            if (!hi) tile[col][col] = BIG;
        }
        __syncthreads();

        // Selection: each lane scans 8 candidates of its row.
#pragma unroll
        for (int cc = 0; cc < 8; ++cc) {
            float d = tile[myrow][cbase + cc];
            if (d < bd[Kk - 1]) {               // threshold guard (common path)
                float pd = d;
                int   pi = m0 + cbase + cc;     // branchless bubble insert
#pragma unroll
                for (int t = 0; t < Kk; ++t) {
                    bool  sw = pd < bd[t];
                    float nd = sw ? bd[t] : pd;
                    int   ni = sw ? bi[t] : pi;
                    bd[t]    = sw ? pd : bd[t];
                    bi[t]    = sw ? pi : bi[t];
                    pd = nd; pi = ni;
                }
            }
        }
        __syncthreads();

        // rotate pipeline registers
        bx = nbx; by = nby; bz = nbz; sc = nsc;
    }

    // Merge the two sorted half-lists of each row (two-pointer, take 16).
    const int slot = hi ? 16 : 0;
#pragma unroll
    for (int k = 0; k < Kk; ++k) {
        md[myrow][slot + k] = bd[k];
        mi[myrow][slot + k] = bi[k];
    }
    __syncthreads();
    if (!hi) {
        int ia = 0, ib = 16;
        int* out = knn + ((size_t)b * Nn + n0 + myrow) * Kk;
        for (int k = 0; k < Kk; ++k) {
            float da = md[myrow][ia];
            float db = md[myrow][ib];
            bool  ta = da <= db;
            out[k] = ta ? mi[myrow][ia] : mi[myrow][ib];
            ia += ta ? 1 : 0;
            ib += ta ? 0 : 1;
        }
    }
}

// ---------------------------------------------------------------------------
// Kernel 3: per-edge |‖pr_n - pr_j‖ - ‖p_n - p_j‖| with block-tree reduction.
// Grid is exactly B*N*K / 256 blocks.
// ---------------------------------------------------------------------------
__global__ void edge_kernel(const float* __restrict__ pref,
                            const float* __restrict__ pts,
                            const int* __restrict__ knn,
                            float* __restrict__ partials) {
    __shared__ float red[256];
    int tid = threadIdx.x;
    int e   = blockIdx.x * 256 + tid;            // edge id in [0, B*N*K)

    int n = (e >> 4) & (Nn - 1);
    int b = e >> 17;                             // 4 (K) + 13 (N) bits
    int j = knn[e];

    const float* prn = pref + ((size_t)b * Nn + n) * 3;
    const float* prj = pref + ((size_t)b * Nn + j) * 3;
    const float* pn  = pts  + ((size_t)b * Nn + n) * 3;
    const float* pj  = pts  + ((size_t)b * Nn + j) * 3;

    float dx = prn[0] - prj[0], dy = prn[1] - prj[1], dz = prn[2] - prj[2];
    float dref = sqrtf(dx * dx + dy * dy + dz * dz);
    dx = pn[0] - pj[0]; dy = pn[1] - pj[1]; dz = pn[2] - pj[2];
    float dpp = sqrtf(dx * dx + dy * dy + dz * dz);

    red[tid] = fabsf(dref - dpp);
    __syncthreads();
    for (int s = 128; s > 0; s >>= 1) {
        if (tid < s) red[tid] += red[tid + s];
        __syncthreads();
    }
    if (tid == 0) partials[blockIdx.x] = red[0];
}

// ---------------------------------------------------------------------------
// Kernel 4: deterministic final reduction of 1024 partials -> mean
// ---------------------------------------------------------------------------
__global__ void final_kernel(const float* __restrict__ partials,
                             float* __restrict__ out) {
    __shared__ float red[256];
    int tid = threadIdx.x;
    float v = partials[tid] + partials[tid + 256] +
              partials[tid + 512] + partials[tid + 768];
    red[tid] = v;
    __syncthreads();
    for (int s = 128; s > 0; s >>= 1) {
        if (tid < s) red[tid] += red[tid + s];
        __syncthreads();
    }
    if (tid == 0) out[0] = red[0] * (1.0f / (float)(Bb * Nn * Kk));
}

// ---------------------------------------------------------------------------
extern "C" void kernel_launch(void* const* d_in, const int* in_sizes, int n_in,
                              void* d_out, int out_size, void* d_ws, size_t ws_size,
                              hipStream_t stream) {
    const float* pref = (const float*)d_in[0];   // points_ref [B,N,3] f32
    const float* pts  = (const float*)d_in[1];   // points     [B,N,3] f32
    float* out = (float*)d_out;                  // scalar f32

    char* ws = (char*)d_ws;
    float* sq       = (float*)ws;                                     // B*N f32
    int*   knn      = (int*)(ws + (size_t)Bb * Nn * sizeof(float));   // B*N*K i32
    float* partials = (float*)(ws + (size_t)Bb * Nn * sizeof(float)
                                  + (size_t)Bb * Nn * Kk * sizeof(int)); // 1024 f32

    sq_kernel<<<(Bb * Nn + 255) / 256, 256, 0, stream>>>(pref, sq);
    knn_kernel<<<dim3(Nn / 16, Bb), 32, 0, stream>>>(pref, sq, knn);
    edge_kernel<<<(Bb * Nn * Kk) / 256, 256, 0, stream>>>(pref, pts, knn, partials);
    final_kernel<<<1, 256, 0, stream>>>(partials, out);
}